// ProtoNet_56521769615978
// MI455X (gfx1250) — compile-verified
//
#include <hip/hip_runtime.h>

typedef float v2f __attribute__((ext_vector_type(2)));
typedef float v4f __attribute__((ext_vector_type(4)));
typedef float v8f __attribute__((ext_vector_type(8)));

#define BATCH   4
#define NWAY    64
#define NSHOT   5
#define DIM     512
#define NQ      960
#define SCALE_CLS 10.0f

// ---------------------------------------------------------------------------
// Kernel 1: prototypes = mean over shots, plus ||proto||^2 per (b, way).
// One block per (b, way); 128 threads, each owning 4 consecutive d's.
// ---------------------------------------------------------------------------
__global__ __launch_bounds__(128) void proto_kernel(
    const float* __restrict__ x_shot,   // [B, NWAY, NSHOT, DIM]
    float* __restrict__ protos,         // [B, NWAY, DIM]
    float* __restrict__ pnorm)          // [B, NWAY]
{
    const int bw   = blockIdx.x;        // 0 .. B*NWAY-1
    const int t    = threadIdx.x;       // 0 .. 127
    const float* src = x_shot + (size_t)bw * (NSHOT * DIM) + t * 4;

    v4f s = {};
#pragma unroll
    for (int sh = 0; sh < NSHOT; ++sh) {
        v4f v = *(const v4f*)(src + sh * DIM);
        s += v;
    }
    s *= 0.2f;   // mean over 5 shots
    *(v4f*)(protos + (size_t)bw * DIM + t * 4) = s;

    float p = s[0]*s[0] + s[1]*s[1] + s[2]*s[2] + s[3]*s[3];
    // wave32 butterfly reduction
#pragma unroll
    for (int off = 16; off > 0; off >>= 1)
        p += __shfl_xor(p, off, 32);

    __shared__ float red[4];
    const int wave = t >> 5, lane = t & 31;
    if (lane == 0) red[wave] = p;
    __syncthreads();
    if (t == 0) pnorm[bw] = red[0] + red[1] + red[2] + red[3];
}

// ---------------------------------------------------------------------------
// Kernel 2: ||query||^2. One wave32 per query row; 8 rows per 256-thread block.
// ---------------------------------------------------------------------------
__global__ __launch_bounds__(256) void qnorm_kernel(
    const float* __restrict__ x_query,  // [B, NQ, DIM] flattened rows
    float* __restrict__ qnorm)          // [B*NQ]
{
    const int row  = blockIdx.x * 8 + (threadIdx.x >> 5);
    const int lane = threadIdx.x & 31;
    const float* src = x_query + (size_t)row * DIM;

    float p = 0.f;
#pragma unroll
    for (int i = 0; i < 4; ++i) {
        v4f v = *(const v4f*)(src + i * 128 + lane * 4);
        p += v[0]*v[0] + v[1]*v[1] + v[2]*v[2] + v[3]*v[3];
    }
#pragma unroll
    for (int off = 16; off > 0; off >>= 1)
        p += __shfl_xor(p, off, 32);
    if (lane == 0) qnorm[row] = p;
}

// ---------------------------------------------------------------------------
// Kernel 3: WMMA fp32 GEMM  dot[b,q,w] = sum_d Q[b,q,d] * P[b,w,d]
// fused with the distance epilogue:
//   out = SCALE * (2*dot - qn - pn)
//
// Block: 256 threads = 8 wave32s. Each wave computes one 16x16 C tile via
// v_wmma_f32_16x16x4_f32 over K=512 (128 steps of K=4).
// Block covers 32 queries x 64 ways: wave (mi,ni) with mi in [0,2), ni in [0,4).
//
// A (16x4 f32) VGPR layout: lane L -> row M = L&15, K-pair base 2*(L>>4);
//   a[0] = A[M][k+base], a[1] = A[M][k+base+1]  -> one float2 load.
// B (4x16 f32): lane L -> col N = L&15, same K-pair striping; since
//   B[k][n] = P[n][k], the operand is a float2 from the prototype row.
// C/D (16x16 f32): VGPR r -> M = r + 8*(L>=16), N = L&15.
// ---------------------------------------------------------------------------
__global__ __launch_bounds__(256) void gemm_dist_kernel(
    const float* __restrict__ x_query,  // [B, NQ, DIM]
    const float* __restrict__ protos,   // [B, NWAY, DIM]
    const float* __restrict__ qnorm,    // [B*NQ]
    const float* __restrict__ pnorm,    // [B*NWAY]
    float* __restrict__ out)            // [B, NQ, NWAY]
{
    const int blk    = blockIdx.x;          // 0 .. B*30-1
    const int b      = blk / (NQ / 32);
    const int qblock = (blk % (NQ / 32)) * 32;

    const int wid  = threadIdx.x >> 5;      // 0..7
    const int lane = threadIdx.x & 31;
    const int mi   = wid >> 2;              // 0..1  (M sub-tile)
    const int ni   = wid & 3;               // 0..3  (N sub-tile)

    const int qtile = qblock + mi * 16;     // first query row of this tile
    const int wtile = ni * 16;              // first way of this tile

    const int mn    = lane & 15;            // row of A / col of B
    const int khalf = (lane >> 4) * 2;      // K-pair base within each K=4 step

    const float* qptr = x_query + ((size_t)b * NQ   + qtile + mn) * DIM + khalf;
    const float* pptr = protos  + ((size_t)b * NWAY + wtile + mn) * DIM + khalf;

    v8f acc = {};
#pragma unroll 8
    for (int k = 0; k < DIM; k += 4) {
        v2f a  = *(const v2f*)(qptr + k);
        v2f bb = *(const v2f*)(pptr + k);
        acc = __builtin_amdgcn_wmma_f32_16x16x4_f32(
            /*neg_a=*/false, a, /*neg_b=*/false, bb,
            /*c_mod=*/(short)0, acc, /*reuse_a=*/false, /*reuse_b=*/false);
    }

    // Epilogue: expanded squared L2 distance, scaled.
    const int   moff = (lane >> 4) * 8;
    const float pnv  = pnorm[b * NWAY + wtile + mn];
    const float* qn  = qnorm + (size_t)b * NQ + qtile;
    float* o = out + ((size_t)b * NQ) * NWAY;

#pragma unroll
    for (int r = 0; r < 8; ++r) {
        const int m = r + moff;
        const float v = SCALE_CLS * (2.f * acc[r] - qn[m] - pnv);
        o[(size_t)(qtile + m) * NWAY + (wtile + mn)] = v;
    }
}

// ---------------------------------------------------------------------------
extern "C" void kernel_launch(void* const* d_in, const int* in_sizes, int n_in,
                              void* d_out, int out_size, void* d_ws, size_t ws_size,
                              hipStream_t stream) {
    const float* x_shot  = (const float*)d_in[0];  // [4, 64, 5, 512]
    const float* x_query = (const float*)d_in[1];  // [4, 960, 512]
    float* out = (float*)d_out;                    // [4, 960, 64]

    float* protos = (float*)d_ws;                  // B*NWAY*DIM floats
    float* pnorm  = protos + (size_t)BATCH * NWAY * DIM;   // B*NWAY
    float* qnorm  = pnorm  + (size_t)BATCH * NWAY;         // B*NQ

    proto_kernel<<<BATCH * NWAY, 128, 0, stream>>>(x_shot, protos, pnorm);
    qnorm_kernel<<<(BATCH * NQ) / 8, 256, 0, stream>>>(x_query, qnorm);
    gemm_dist_kernel<<<BATCH * (NQ / 32), 256, 0, stream>>>(
        x_query, protos, qnorm, pnorm, out);
}